// LowEmb_33878702031228
// MI455X (gfx1250) — compile-verified
//
#include <hip/hip_runtime.h>
#include <hip/hip_bf16.h>

// ---------------------------------------------------------------------------
// t-SNE KL(P||Q) for N=8192, d_low=2.
//   Pass 1 (compute-bound): S = sum_{i!=j} (1 + |Yi-Yj|^2/dof)^(-(dof+1)/2)
//       16x16 tiles of the Gram matrix via V_WMMA_F32_16X16X4_F32 (K=2 padded
//       to 4 with zeros; lanes 16..31 of A and B carry the zero pad).
//       dof==1 fast loop (raw v_rcp_f32) split from the generic __powf loop
//       so the hot loop body stays tight.
//   Pass 2 (bandwidth-bound, 256 MB of P @ 23.3 TB/s ~ 11 us): streaming
//       nontemporal float4 loads of P, per element
//       term = P*(log P - max(expo*log(1+D/dof) - log S, log EPS)),
//       diagonal forced to Qn = EPS as in the reference.
//   Pass 3: scalar finalize to d_out.
// Scratch: d_ws[0..1] = { S (double), KL (double) }, zeroed via memset node.
// ---------------------------------------------------------------------------

#define NPTS   8192
#define NTILE  (NPTS / 16)             // 512 16x16 tiles per side
#define LOGEPS (-27.631021115928547f)  // ln(1e-12)

typedef float v2f __attribute__((ext_vector_type(2)));
typedef float v4f __attribute__((ext_vector_type(4)));
typedef float v8f __attribute__((ext_vector_type(8)));

__device__ __forceinline__ void block_reduce_add(double v, double* target) {
    __shared__ double red[256];
    red[threadIdx.x] = v;
    __syncthreads();
    for (int s = 128; s > 0; s >>= 1) {
        if ((int)threadIdx.x < s) red[threadIdx.x] += red[threadIdx.x + s];
        __syncthreads();
    }
    if (threadIdx.x == 0) atomicAdd(target, red[0]);
}

// ----------------------------- Pass 1: S ----------------------------------
// One block per 16-row strip; 8 waves sweep the 512 column tiles.
// Each wave: 1 v_wmma_f32_16x16x4_f32 -> 256 dot products per instruction.
__global__ __launch_bounds__(256) void
sumq_kernel(const float2* __restrict__ Y, const int* __restrict__ dofp,
            double* __restrict__ ws_S) {
    const int lane = threadIdx.x & 31;
    const int wave = threadIdx.x >> 5;
    const int n    = lane & 15;          // column within tile owned by lane
    const bool lo  = lane < 16;
    const int i0   = blockIdx.x * 16;

    const int   dof    = dofp[0];
    const float invdof = 1.0f / (float)dof;
    const float expo   = -0.5f * (float)(dof + 1);

    // A-matrix (16x4, K=2..3 zero pad): lanes 0-15 hold K=0,1; 16-31 hold pad.
    const float2 yi  = Y[i0 + n];
    const float  sqi = yi.x * yi.x + yi.y * yi.y;
    v2f a; a.x = lo ? yi.x : 0.0f; a.y = lo ? yi.y : 0.0f;

    // Row norms (+1 pre-folded) for the 8 C/D rows this lane owns
    // (m = r + (lo?0:8)); lane m holds |Y[i0+m]|^2, broadcast via shuffle.
    float yi2p1[8];
#pragma unroll
    for (int r = 0; r < 8; ++r)
        yi2p1[r] = __shfl(sqi, r + (lo ? 0 : 8), 32) + 1.0f;
    const int rowbase = i0 + (lo ? 0 : 8);

    double acc = 0.0;

    if (dof == 1) {
        // ---- hot loop: q = 1/(1 + D), raw v_rcp_f32 ----
        for (int jt = wave; jt < NTILE; jt += 8) {
            const int j0 = jt * 16;
            const float2 yj  = Y[j0 + n];
            const float  sqj = yj.x * yj.x + yj.y * yj.y;
            v2f b; b.x = lo ? yj.x : 0.0f; b.y = lo ? yj.y : 0.0f;

            v8f c = {0.f, 0.f, 0.f, 0.f, 0.f, 0.f, 0.f, 0.f};
            c = __builtin_amdgcn_wmma_f32_16x16x4_f32(
                    false, a, false, b, (short)0, c, false, false);

            float tsum = 0.0f;
#pragma unroll
            for (int r = 0; r < 8; ++r) {
                // 1 + D = (1 + |yi|^2 + |yj|^2) - 2*yi.yj
                const float base = fmaf(-2.0f, c[r], yi2p1[r] + sqj);
                float q = __builtin_amdgcn_rcpf(base);
                if (rowbase + r == j0 + n) q = 0.0f;   // zero diagonal
                tsum += q;
            }
            acc += (double)tsum;
        }
    } else {
        // ---- generic dof: q = (1 + D/dof)^(-(dof+1)/2) ----
        for (int jt = wave; jt < NTILE; jt += 8) {
            const int j0 = jt * 16;
            const float2 yj  = Y[j0 + n];
            const float  sqj = yj.x * yj.x + yj.y * yj.y;
            v2f b; b.x = lo ? yj.x : 0.0f; b.y = lo ? yj.y : 0.0f;

            v8f c = {0.f, 0.f, 0.f, 0.f, 0.f, 0.f, 0.f, 0.f};
            c = __builtin_amdgcn_wmma_f32_16x16x4_f32(
                    false, a, false, b, (short)0, c, false, false);

            float tsum = 0.0f;
#pragma unroll
            for (int r = 0; r < 8; ++r) {
                const float Dm   = fmaf(-2.0f, c[r], yi2p1[r] + sqj) - 1.0f;
                const float base = fmaf(Dm, invdof, 1.0f);
                float q = __powf(base, expo);
                if (rowbase + r == j0 + n) q = 0.0f;   // zero diagonal
                tsum += q;
            }
            acc += (double)tsum;
        }
    }
    block_reduce_add(acc, ws_S);
}

// ----------------------------- Pass 2: KL ---------------------------------
// One block per row i; coalesced nontemporal float4 stream of P (read-once),
// Y stays resident in WGP$/L2 (64 KB).
__global__ __launch_bounds__(256) void
kl_kernel(const float* __restrict__ P, const float2* __restrict__ Y,
          const int* __restrict__ dofp, const double* __restrict__ ws_S,
          double* __restrict__ ws_KL) {
    const int i = blockIdx.x;
    const float2 yi = Y[i];

    const int   dof    = dofp[0];
    const float invdof = 1.0f / (float)dof;
    const float expo   = -0.5f * (float)(dof + 1);
    const float logS   = __logf((float)ws_S[0]);   // pass 1 done (stream order)

    const v4f* __restrict__ P4 = (const v4f*)(P + (size_t)i * NPTS);
    const v4f* __restrict__ Y4 = (const v4f*)Y;    // {Y[2k].x,.y, Y[2k+1].x,.y}

    double acc = 0.0;
    for (int jb = threadIdx.x * 4; jb < NPTS; jb += 256 * 4) {
        const v4f p4 = __builtin_nontemporal_load(&P4[jb >> 2]);
        const v4f ya = Y4[jb >> 1];
        const v4f yb = Y4[(jb >> 1) + 1];
        const float yx[4] = {ya[0], ya[2], yb[0], yb[2]};
        const float yy[4] = {ya[1], ya[3], yb[1], yb[3]};
#pragma unroll
        for (int k = 0; k < 4; ++k) {
            const int   j  = jb + k;
            const float p  = p4[k];
            const float dx = yi.x - yx[k];
            const float dy = yi.y - yy[k];
            const float Dm = dx * dx + dy * dy;
            float logQn;
            if (j == i) {
                logQn = LOGEPS;                              // Q[i,i]=0 -> EPS
            } else {
                // log(q/S) = expo*log(1+D/dof) - log S, clamped at log EPS
                logQn = fmaxf(expo * __logf(fmaf(Dm, invdof, 1.0f)) - logS,
                              LOGEPS);
            }
            acc += (double)(p * (__logf(p) - logQn));
        }
    }
    block_reduce_add(acc, ws_KL);
}

// ----------------------------- Pass 3 -------------------------------------
__global__ void finalize_kernel(const double* __restrict__ ws,
                                float* __restrict__ out) {
    if (threadIdx.x == 0) out[0] = (float)ws[1];
}

extern "C" void kernel_launch(void* const* d_in, const int* in_sizes, int n_in,
                              void* d_out, int out_size, void* d_ws, size_t ws_size,
                              hipStream_t stream) {
    const float*  P   = (const float*)d_in[0];   // (8192, 8192) f32
    const float2* Y   = (const float2*)d_in[1];  // (8192, 2)    f32
    const int*    dof = (const int*)d_in[2];     // scalar int
    double* ws = (double*)d_ws;                  // ws[0]=S, ws[1]=KL

    hipMemsetAsync(ws, 0, 2 * sizeof(double), stream);
    sumq_kernel<<<NTILE, 256, 0, stream>>>(Y, dof, &ws[0]);
    kl_kernel<<<NPTS, 256, 0, stream>>>(P, Y, dof, &ws[0], &ws[1]);
    finalize_kernel<<<1, 32, 0, stream>>>(ws, (float*)d_out);
}